// TAtt_55319178772749
// MI455X (gfx1250) — compile-verified
//
#include <hip/hip_runtime.h>
#include <math.h>

#define B_ 64
#define N_ 1000
#define C_ 64
#define T_ 32

typedef __attribute__((ext_vector_type(2))) float v2f;
typedef __attribute__((ext_vector_type(8))) float v8f;
typedef __attribute__((ext_vector_type(4))) int   v4i;

#if defined(__gfx1250__) && __has_builtin(__builtin_amdgcn_global_load_async_to_lds_b128)
#define HAVE_ASYNC_LDS 1
typedef __attribute__((address_space(1))) v4i* as1_v4i_ptr;  // global v4i*
typedef __attribute__((address_space(3))) v4i* as3_v4i_ptr;  // LDS v4i*
#if __has_builtin(__builtin_amdgcn_s_wait_asynccnt)
#define WAIT_ASYNC(n) __builtin_amdgcn_s_wait_asynccnt(n)
#else
#define WAIT_ASYNC(n) asm volatile("s_wait_asynccnt %0" ::"i"(n) : "memory")
#endif
// one lane: 16B global -> 16B LDS, tracked on ASYNCcnt
__device__ __forceinline__ void async_cp16(const float* g, float* lds) {
    __builtin_amdgcn_global_load_async_to_lds_b128(
        (as1_v4i_ptr)(g), (as3_v4i_ptr)(lds), 0, 0);
}
#else
#define HAVE_ASYNC_LDS 0
#endif

// ---------------- zero scratch (lhs_pre is atomically accumulated) ----------
__global__ void k_zero(float* __restrict__ p, int n) {
    int i = blockIdx.x * blockDim.x + threadIdx.x;
    if (i < n) p[i] = 0.f;
}

// ---------------- pass 1 over X: rhs[b,n,t] and lhs_pre[b,t,c] --------------
// block = (b, 8 consecutive n). X tile (C=64,T=32) staged in LDS (stride 36,
// conflict-free for both row and column access patterns).
__global__ __launch_bounds__(256)
void k_reduce(const float* __restrict__ X, const float* __restrict__ U1,
              const float* __restrict__ U3, float* __restrict__ lhs_pre,
              float* __restrict__ rhs) {
    __shared__ float Xs[C_ * 36];
    __shared__ float u3s[C_];
    __shared__ float red[8 * T_];
    const int tid = threadIdx.x;
    const int blocksPerB = N_ / 8;                 // 125
    const int b  = blockIdx.x / blocksPerB;
    const int n0 = (blockIdx.x % blocksPerB) * 8;
    if (tid < C_) u3s[tid] = U3[tid];

    // per-thread staging coordinates: two float4s per tile
    const int p0 = tid, p1 = tid + 256;
    const int l0 = ((p0 * 4) >> 5) * 36 + ((p0 * 4) & 31);
    const int l1 = ((p1 * 4) >> 5) * 36 + ((p1 * 4) & 31);

    float lacc[8];
#pragma unroll
    for (int j = 0; j < 8; ++j) lacc[j] = 0.f;

    const int t_ = tid & 31;   // t for rhs reduction
    const int cg = tid >> 5;   // c-group 0..7

    for (int k = 0; k < 8; ++k) {
        const int n = n0 + k;
        const float* src = X + ((size_t)(b * N_ + n)) * C_ * T_;
        __syncthreads();                            // Xs / red reuse guard
#if HAVE_ASYNC_LDS
        async_cp16(src + p0 * 4, &Xs[l0]);          // global -> LDS, no VGPR trip
        async_cp16(src + p1 * 4, &Xs[l1]);
        WAIT_ASYNC(0);
#else
        {
            const float4* s4 = (const float4*)src;
            float4 v0 = s4[p0];
            float4 v1 = s4[p1];
            *(float4*)&Xs[l0] = v0;
            *(float4*)&Xs[l1] = v1;
        }
#endif
        __syncthreads();

        // rhs[b,n,t] = sum_c X[c,t] * U3[c]  (8-way split over c, LDS reduce)
        float s = 0.f;
#pragma unroll
        for (int c = cg; c < C_; c += 8) s += Xs[c * 36 + t_] * u3s[c];
        red[cg * T_ + t_] = s;

        // lhs_pre[b,t,c] += U1[n] * X[c,t]  (register accumulation over n)
        float u1 = U1[n];
#pragma unroll
        for (int j = 0; j < 8; ++j) {
            int e  = tid * 8 + j;     // flat (t,c), t = e/64, c = e%64
            int tt = e >> 6;
            int cc = e & 63;
            lacc[j] += u1 * Xs[cc * 36 + tt];
        }
        __syncthreads();
        if (tid < T_) {
            float acc = 0.f;
#pragma unroll
            for (int j = 0; j < 8; ++j) acc += red[j * T_ + tid];
            rhs[((size_t)(b * N_ + n)) * T_ + tid] = acc;
        }
    }
#pragma unroll
    for (int j = 0; j < 8; ++j) {
        int e  = tid * 8 + j;
        int tt = e >> 6;
        int cc = e & 63;
        atomicAdd(&lhs_pre[(b * T_ + tt) * C_ + cc], lacc[j]);
    }
}

// ---------------- lhs[b,t,n] = lhs_pre[b,t,:] @ U2[:,n] ---------------------
__global__ __launch_bounds__(128)
void k_lhs(const float* __restrict__ lhs_pre, const float* __restrict__ U2,
           float* __restrict__ lhs) {
    __shared__ float lp[C_];
    const int bt  = blockIdx.x;     // b*T + t
    const int tid = threadIdx.x;
    if (tid < C_) lp[tid] = lhs_pre[bt * C_ + tid];
    __syncthreads();
    for (int i = 0; i < 8; ++i) {
        int n = tid + i * 128;
        if (n < N_) {
            float acc = 0.f;
#pragma unroll
            for (int c = 0; c < C_; ++c) acc += lp[c] * U2[c * N_ + n];
            lhs[(size_t)bt * N_ + n] = acc;
        }
    }
}

// ---------------- per-b: scores -> sigmoid -> Ve@S -> softmax(axis=1) -------
__global__ __launch_bounds__(1024)
void k_scores(const float* __restrict__ lhs, const float* __restrict__ rhs,
              const float* __restrict__ be, const float* __restrict__ Ve,
              float* __restrict__ E) {
    __shared__ float Ss[T_ * 33];
    __shared__ float Eo[T_ * 33];
    __shared__ float mx[T_], inv[T_];
    const int b   = blockIdx.x;
    const int tid = threadIdx.x;
    const int t = tid >> 5;
    const int s = tid & 31;

    const float* lr = lhs + ((size_t)b * T_ + t) * N_;   // broadcast per 32 lanes
    const float* rr = rhs + (size_t)b * N_ * T_ + s;     // coalesced per wave
    float sc = be[t * T_ + s];
    for (int n = 0; n < N_; ++n) sc += lr[n] * rr[(size_t)n * T_];
    Ss[t * 33 + s] = 1.f / (1.f + __expf(-sc));          // sigmoid(scores)[t][s]
    __syncthreads();

    float e0 = 0.f;                                      // E0[t,u] = Ve @ S
#pragma unroll
    for (int k = 0; k < T_; ++k) e0 += Ve[t * T_ + k] * Ss[k * 33 + s];
    Eo[t * 33 + s] = e0;
    __syncthreads();

    if (tid < T_) {                                      // softmax over t (col u=tid)
        float m = -1e30f;
        for (int k = 0; k < T_; ++k) m = fmaxf(m, Eo[k * 33 + tid]);
        float sum = 0.f;
        for (int k = 0; k < T_; ++k) sum += __expf(Eo[k * 33 + tid] - m);
        mx[tid]  = m;
        inv[tid] = 1.f / sum;
    }
    __syncthreads();
    E[((size_t)b * T_ + t) * T_ + s] = __expf(Eo[t * 33 + s] - mx[s]) * inv[s];
}

// ---------------- pass 2 over X: X_hat[b,n] = X[b,n](64x32) @ E[b](32x32) ---
// 256 threads = 8 waves; wave w owns output tile (c: 16*(w&3), s: 16*(w>>2)).
// 8 x V_WMMA_F32_16X16X4_F32 per tile. X tiles double-buffered in LDS with
// async global->LDS loads overlapping the WMMA work on the previous tile.
__global__ __launch_bounds__(256)
void k_apply(const float* __restrict__ X, const float* __restrict__ E,
             float* __restrict__ out) {
#if HAVE_ASYNC_LDS
    __shared__ float Xs[2][C_ * 36];
#else
    __shared__ float Xs[1][C_ * 36];
#endif
    __shared__ float Es[T_ * 48];
    const int tid = threadIdx.x;
    const int grpPerB = N_ / 4;                    // 250
    const int b  = blockIdx.x / grpPerB;
    const int n0 = (blockIdx.x % grpPerB) * 4;

    {   // stage E[b] (1024 floats) into LDS, stride 48 (conflict-free B frags)
        const float4* eg = (const float4*)(E + (size_t)b * T_ * T_);
        float4 v = eg[tid];
        int flat = tid * 4;
        int t = flat >> 5;
        int u = flat & 31;
        *(float4*)&Es[t * 48 + u] = v;
    }

    // per-thread staging coordinates: two float4s per tile
    const int p0 = tid, p1 = tid + 256;
    const int l0 = ((p0 * 4) >> 5) * 36 + ((p0 * 4) & 31);
    const int l1 = ((p1 * 4) >> 5) * 36 + ((p1 * 4) & 31);

    const int lane = tid & 31;
    const int wave = tid >> 5;
    const int rt = (wave & 3) * 16;   // output row tile base (c)
    const int qt = (wave >> 2) * 16;  // output col tile base (s)
    const int mh = lane & 15;
    const int kh = (lane >> 4) * 2;   // K half-wave offset per ISA A/B layout

#if HAVE_ASYNC_LDS
    {   // prologue: tile n0 -> buffer 0
        const float* src = X + ((size_t)(b * N_ + n0)) * C_ * T_;
        async_cp16(src + p0 * 4, &Xs[0][l0]);
        async_cp16(src + p1 * 4, &Xs[0][l1]);
    }
#endif

    for (int nn = 0; nn < 4; ++nn) {
        const int n = n0 + nn;
#if HAVE_ASYNC_LDS
        const int cur = nn & 1;
        if (nn + 1 < 4) {   // issue next tile into the other buffer
            const float* src = X + ((size_t)(b * N_ + n + 1)) * C_ * T_;
            async_cp16(src + p0 * 4, &Xs[1 - cur][l0]);
            async_cp16(src + p1 * 4, &Xs[1 - cur][l1]);
            WAIT_ASYNC(2);  // only next tile's 2 ops may remain in flight
        } else {
            WAIT_ASYNC(0);
        }
        __syncthreads();    // tile n resident + visible to all waves
#else
        const int cur = 0;
        const float* src = X + ((size_t)(b * N_ + n)) * C_ * T_;
        __syncthreads();                           // Xs reuse guard
        {
            const float4* s4 = (const float4*)src;
            float4 v0 = s4[p0];
            float4 v1 = s4[p1];
            *(float4*)&Xs[0][l0] = v0;
            *(float4*)&Xs[0][l1] = v1;
        }
        const int npf = (n + 1 < N_) ? n + 1 : n;
        __builtin_prefetch(X + ((size_t)(b * N_ + npf)) * C_ * T_ + tid * 8, 0, 0);
        __syncthreads();
#endif

        v8f acc = {};
#pragma unroll
        for (int k0 = 0; k0 < T_; k0 += 4) {
            v2f a, bb;
            a.x  = Xs[cur][(rt + mh) * 36 + k0 + kh];   // A 16x4: lane=M, half=K
            a.y  = Xs[cur][(rt + mh) * 36 + k0 + kh + 1];
            bb.x = Es[(k0 + kh) * 48 + qt + mh];        // B 4x16: lane=N, half=K
            bb.y = Es[(k0 + kh + 1) * 48 + qt + mh];
            acc = __builtin_amdgcn_wmma_f32_16x16x4_f32(
                false, a, false, bb, (short)0, acc, false, false);
        }

        float* op = out + ((size_t)(b * N_ + n)) * C_ * T_;
#pragma unroll
        for (int i = 0; i < 8; ++i) {                   // D: VGPR i -> M=i / 8+i
            int c = rt + i + 8 * (lane >> 4);
            op[c * T_ + qt + mh] = acc[i];
        }
        __syncthreads();    // everyone done with Xs[cur] before it is refilled
    }
}

extern "C" void kernel_launch(void* const* d_in, const int* in_sizes, int n_in,
                              void* d_out, int out_size, void* d_ws, size_t ws_size,
                              hipStream_t stream) {
    const float* X  = (const float*)d_in[0];
    const float* U1 = (const float*)d_in[1];
    const float* U2 = (const float*)d_in[2];
    const float* U3 = (const float*)d_in[3];
    const float* be = (const float*)d_in[4];
    const float* Ve = (const float*)d_in[5];
    float* out = (float*)d_out;

    float* ws      = (float*)d_ws;
    float* lhs_pre = ws;                               // B*T*C   = 131072
    float* rhs     = lhs_pre + (size_t)B_ * T_ * C_;   // B*N*T   = 2048000
    float* lhs     = rhs + (size_t)B_ * N_ * T_;       // B*T*N   = 2048000
    float* E       = lhs + (size_t)B_ * T_ * N_;       // B*T*T   = 65536

    k_zero<<<(B_ * T_ * C_ + 255) / 256, 256, 0, stream>>>(lhs_pre, B_ * T_ * C_);
    k_reduce<<<B_ * (N_ / 8), 256, 0, stream>>>(X, U1, U3, lhs_pre, rhs);
    k_lhs<<<B_ * T_, 128, 0, stream>>>(lhs_pre, U2, lhs);
    k_scores<<<B_, 1024, 0, stream>>>(lhs, rhs, be, Ve, E);
    k_apply<<<B_ * (N_ / 4), 256, 0, stream>>>(X, E, out);
}